// ROIAlign_80796924773064
// MI455X (gfx1250) — compile-verified
//
#include <hip/hip_runtime.h>
#include <stdint.h>

// ---------------- problem constants (from reference) ----------------
#define K_      7          // KSIZE
#define SR_     2          // sampling ratio
#define H_      64
#define W_      64
#define C_      256
#define NBINS   (K_ * K_)  // 49

#define WAVES_PER_BLOCK 4
#define ROWS        16                       // 4 samples x 4 bilinear corners
#define ROW_BYTES   (C_ * 4)                 // 1024 B per gathered feature row
#define TILE_BYTES  (ROWS * ROW_BYTES)       // 16 KB per gather
#define SLICE_BYTES (2 * TILE_BYTES)         // per-wave double buffer, 32 KB
#define LDS_BYTES   (WAVES_PER_BLOCK * SLICE_BYTES) // 128 KB < 320 KB/WGP

typedef unsigned int v4u __attribute__((ext_vector_type(4)));
typedef int          v8i __attribute__((ext_vector_type(8)));
typedef int          v4i __attribute__((ext_vector_type(4)));
typedef float        vf4 __attribute__((ext_vector_type(4)));

// Only meaningful on the device pass; on the host pass AMDGPU builtins are
// reached via the aux-target mechanism and __has_builtin reports false.
#if defined(__HIP_DEVICE_COMPILE__)
#if !__has_builtin(__builtin_amdgcn_tensor_load_to_lds)
#error "gfx1250 TDM builtin __builtin_amdgcn_tensor_load_to_lds not available"
#endif
#endif

// LDS byte offset of a generic pointer known to point into __shared__.
// AS(3) pointers are 32-bit offsets from the wave's LDS base in the AMDGPU
// data layout, which is exactly what the TDM descriptor's lds_addr wants.
typedef __attribute__((address_space(3))) const char lds_cchar_t;
static __device__ __forceinline__ uint32_t lds_addr_of(const void* p) {
    return (uint32_t)(uintptr_t)(lds_cchar_t*)p;
}

static __device__ __forceinline__ uint32_t rfl(uint32_t v) {
    return (uint32_t)__builtin_amdgcn_readfirstlane((int)v);
}

// Issue one TDM gather: 16 rows (16-bit indices, each row C_ f32 elements)
// from the 2D tensor [H*W, C] at gaddr into LDS at lds_byte_off.
// 6-arg builtin form: groups 0..3 of the D#, plus the unused VADDR4 slot
// passed as a zero int32x8, then cpol.
static __device__ __forceinline__ void tdm_gather16(uint32_t lds_byte_off,
                                                    uint64_t gaddr,
                                                    const uint32_t pk[8]) {
    v4u g0;
    g0[0] = 0x80000001u;                       // count=1, gather_mode=1, 16-bit idx
    g0[1] = rfl(lds_byte_off);                 // lds_addr (bytes)
    g0[2] = rfl((uint32_t)gaddr);              // global_addr[31:0]
    g0[3] = (rfl((uint32_t)(gaddr >> 32)) & 0x01ffffffu) | (2u << 30); // type=2

    v8i g1;
    g1[0] = (int)(2u << 16);                   // workgroup_mask=0, data_size=4B
    g1[1] = (int)((uint32_t)C_ << 16);         // tensor_dim0 = 256 elements
    g1[2] = (int)((uint32_t)(H_ * W_) << 16);  // tensor_dim1 = 4096 rows
    g1[3] = (int)((uint32_t)C_ << 16);         // tile_dim0  = 256 elements
    g1[4] = ROWS;                              // tile_dim1  = 16 valid indices
    g1[5] = C_;                                // tensor_dim0_stride = 256 elements
    g1[6] = 0;
    g1[7] = 0;

    v4i g2, g3;                                // 16 x 16-bit row indices
    g2[0] = (int)rfl(pk[0]); g2[1] = (int)rfl(pk[1]);
    g2[2] = (int)rfl(pk[2]); g2[3] = (int)rfl(pk[3]);
    g3[0] = (int)rfl(pk[4]); g3[1] = (int)rfl(pk[5]);
    g3[2] = (int)rfl(pk[6]); g3[3] = (int)rfl(pk[7]);

    v8i g4 = {0, 0, 0, 0, 0, 0, 0, 0};         // VADDR4: unused per ISA

    __builtin_amdgcn_tensor_load_to_lds(g0, g1, g2, g3, g4, /*cpol=*/0);
}

// Per-bin geometry: 16 packed row indices + 16 bilinear*pool weights.
static __device__ __forceinline__ void bin_geometry(int bin, float y1, float x1,
                                                    float bin_h, float bin_w,
                                                    uint32_t pk[8], float w[16]) {
    const int ky = bin / K_;
    const int kx = bin - ky * K_;
    int   y0[2], x0[2];
    float ly[2], lx[2];
#pragma unroll
    for (int i = 0; i < 2; ++i) {
        float sy = y1 + ((float)(ky * SR_ + i) + 0.5f) * (1.0f / SR_) * bin_h;
        float cy = fminf(fmaxf(floorf(sy), 0.0f), (float)(H_ - 2));
        y0[i] = (int)cy;
        ly[i] = fminf(fmaxf(sy - cy, 0.0f), 1.0f);

        float sx = x1 + ((float)(kx * SR_ + i) + 0.5f) * (1.0f / SR_) * bin_w;
        float cx = fminf(fmaxf(floorf(sx), 0.0f), (float)(W_ - 2));
        x0[i] = (int)cx;
        lx[i] = fminf(fmaxf(sx - cx, 0.0f), 1.0f);
    }
    uint32_t idx[16];
#pragma unroll
    for (int iy = 0; iy < 2; ++iy) {
#pragma unroll
        for (int ix = 0; ix < 2; ++ix) {
            const int s = iy * 2 + ix;
            const uint32_t base = (uint32_t)(y0[iy] * W_ + x0[ix]);
            idx[s * 4 + 0] = base;
            idx[s * 4 + 1] = base + 1;
            idx[s * 4 + 2] = base + W_;
            idx[s * 4 + 3] = base + W_ + 1;
            const float wy1 = ly[iy], wy0 = 1.0f - wy1;
            const float wx1 = lx[ix], wx0 = 1.0f - wx1;
            w[s * 4 + 0] = 0.25f * wy0 * wx0;   // 0.25 = 2x2 pooling mean
            w[s * 4 + 1] = 0.25f * wy0 * wx1;
            w[s * 4 + 2] = 0.25f * wy1 * wx0;
            w[s * 4 + 3] = 0.25f * wy1 * wx1;
        }
    }
#pragma unroll
    for (int kk = 0; kk < 8; ++kk)
        pk[kk] = idx[2 * kk] | (idx[2 * kk + 1] << 16);
}

__global__ void __launch_bounds__(WAVES_PER_BLOCK * 32)
roialign_tdm_kernel(const float* __restrict__ x, const float* __restrict__ rois,
                    const int* __restrict__ bidx, float* __restrict__ out) {
    extern __shared__ char smem[];
    const int roi  = blockIdx.x;
    const int lane = threadIdx.x & 31;
    const int wave = threadIdx.x >> 5;

    const float y1 = rois[roi * 4 + 0];
    const float x1 = rois[roi * 4 + 1];
    const float y2 = rois[roi * 4 + 2];
    const float x2 = rois[roi * 4 + 3];
    const int   b  = bidx[roi];
    const float bin_h = (y2 - y1) * (1.0f / K_);
    const float bin_w = (x2 - x1) * (1.0f / K_);
    const uint64_t gbase =
        (uint64_t)(uintptr_t)(x + (size_t)b * (size_t)(H_ * W_ * C_));

    char* slice = smem + wave * SLICE_BYTES;           // this wave's buffers
    const uint32_t slice_off = lds_addr_of(slice);

    // wave handles bins: wave, wave+4, wave+8, ...
    const int nb = (NBINS - wave + WAVES_PER_BLOCK - 1) / WAVES_PER_BLOCK;

    uint32_t pk[8];
    float    wcur[16];
    bin_geometry(wave, y1, x1, bin_h, bin_w, pk, wcur);
    tdm_gather16(slice_off, gbase, pk);                // prime buffer 0

    for (int k = 0; k < nb; ++k) {
        const int bin = wave + k * WAVES_PER_BLOCK;
        float wnext[16];
        if (k + 1 < nb) {
            uint32_t pk2[8];
            bin_geometry(bin + WAVES_PER_BLOCK, y1, x1, bin_h, bin_w, pk2, wnext);
            tdm_gather16(slice_off + (uint32_t)(((k + 1) & 1) * TILE_BYTES),
                         gbase, pk2);
            __builtin_amdgcn_s_wait_tensorcnt(1);      // gather(k) complete
        } else {
            __builtin_amdgcn_s_wait_tensorcnt(0);      // drain
        }

        const char* buf = slice + (k & 1) * TILE_BYTES;
        vf4 acc0 = {0.0f, 0.0f, 0.0f, 0.0f};           // channels lane*4 .. +3
        vf4 acc1 = {0.0f, 0.0f, 0.0f, 0.0f};           // channels 128+lane*4 ..
#pragma unroll
        for (int j = 0; j < ROWS; ++j) {
            const vf4* row = (const vf4*)(buf + j * ROW_BYTES);
            vf4 a  = row[lane];
            vf4 bq = row[lane + 32];                   // +512 B = channel 128
            acc0 += a  * wcur[j];
            acc1 += bq * wcur[j];
        }

        float* o = out + ((size_t)roi * NBINS + bin) * C_;
        __builtin_nontemporal_store(acc0, (vf4*)o + lane);          // NT: keep x in L2
        __builtin_nontemporal_store(acc1, (vf4*)(o + C_ / 2) + lane);

        if (k + 1 < nb) {
#pragma unroll
            for (int j = 0; j < 16; ++j) wcur[j] = wnext[j];
        }
    }
}

extern "C" void kernel_launch(void* const* d_in, const int* in_sizes, int n_in,
                              void* d_out, int out_size, void* d_ws, size_t ws_size,
                              hipStream_t stream) {
    const float* x    = (const float*)d_in[0];
    const float* rois = (const float*)d_in[1];
    const int*   bidx = (const int*)d_in[2];
    float*       out  = (float*)d_out;
    const int N = in_sizes[2];                 // one batch_idx entry per ROI

    dim3 grid((unsigned)N);
    dim3 block(WAVES_PER_BLOCK * 32);
    roialign_tdm_kernel<<<grid, block, LDS_BYTES, stream>>>(x, rois, bidx, out);
}